// UniDimModel_16922171146434
// MI455X (gfx1250) — compile-verified
//
#include <hip/hip_runtime.h>

// UniDimModel: feature dim is 1 -> every Linear(1,1) is a scalar FMA; the
// whole graph is elementwise over B = 2^23. Memory-bound: 48 B/elem
// (16 in, 32 out) -> ~402 MB -> ~17 us at 23.3 TB/s HBM.
//
// Round-2 changes:
//  * gfx1250 has a hardware transcendental TANH (ISA §8.4 lists TANH among
//    TRANS ops). Use __builtin_amdgcn_tanhf / _tanh_f32 if clang exposes it
//    (probed via __has_builtin); otherwise a clamped exp2-based tanh
//    (v_exp_f32 is native exp2) -- either way ~1-8 VALU ops instead of the
//    ~35-op OCML tanhf expansion that made round 1 compute-limited.
//  * sigmoid(x) = 0.5 * (1 + tanh(x/2)) reuses the same fast path.
//  * 32-bit per-thread offsets + uniform hoisted stream bases to cut
//    64-bit address arithmetic.

typedef float v4f __attribute__((ext_vector_type(4)));

#define BFW 0.33f
#define LBW 0.33f
#define AREC 0.1f

__device__ __forceinline__ float fast_tanh(float x) {
#if __has_builtin(__builtin_amdgcn_tanhf)
  return __builtin_amdgcn_tanhf(x);          // v_tanh_f32
#elif __has_builtin(__builtin_amdgcn_tanh_f32)
  return __builtin_amdgcn_tanh_f32(x);       // v_tanh_f32
#else
  // tanh(x) = (e^{2x}-1)/(e^{2x}+1); v_exp_f32 computes exp2 natively.
  // Clamp so e never overflows (tanh saturates to +-1 well before |x|=10).
  const float xc = fminf(fmaxf(x, -10.0f), 10.0f);
  const float e = __builtin_amdgcn_exp2f(xc * 2.8853900817779268f); // 2*log2(e)
  return __fdividef(e - 1.0f, e + 1.0f);     // v_rcp_f32 + mul
#endif
}

__device__ __forceinline__ float fast_sigmoid(float x) {
  return fmaf(0.5f, fast_tanh(0.5f * x), 0.5f);
}

struct P {
  float w01, b01, w10, b10, w12, b12, w21, b21, w23, b23, w32, b32;
  float aRecBs, recScale, keep, oneMinusBfw;
};

struct R {
  float outv, a1n, a2n, a3n, rec0, rec1, rec2;
};

__device__ __forceinline__ R compute_one(float a0, float a1, float a2, float a3,
                                         const P& p) {
  R r;
  // Pre-activations (scalar "Linear(1,1)" layers)
  const float z01 = fmaf(a0, p.w01, p.b01);
  const float z10 = fmaf(a1, p.w10, p.b10);
  const float z21 = fmaf(a2, p.w21, p.b21);
  const float z32 = fmaf(a3, p.w32, p.b32);

  // rec = grad_a mean((a*w+b - tgt)^2) = (2/B)*(z - tgt)*w
  r.rec0 = p.recScale * (z10 - a0) * p.w10;
  r.rec1 = p.recScale * (z21 - a1) * p.w21;
  r.rec2 = p.recScale * (z32 - a2) * p.w32;

  // a1n uses old activations only
  r.a1n = BFW * fast_tanh(z01) + p.keep * a1 + LBW * fast_tanh(z21)
        - p.aRecBs * r.rec0;

  // a2n depends on a1n (sequential chain, still elementwise)
  const float z12 = fmaf(r.a1n, p.w12, p.b12);
  r.a2n = BFW * fast_tanh(z12) + p.keep * a2 + LBW * fast_tanh(z32)
        - p.aRecBs * r.rec1;

  const float z23 = fmaf(r.a2n, p.w23, p.b23);
  r.a3n = BFW * fast_tanh(z23) + p.oneMinusBfw * a3 - p.aRecBs * r.rec2;

  r.outv = fast_sigmoid(r.a3n);
  return r;
}

__global__ __launch_bounds__(256) void unidim_kernel(
    const float* __restrict__ a0p, const float* __restrict__ a1p,
    const float* __restrict__ a2p, const float* __restrict__ a3p,
    const float* __restrict__ w01p, const float* __restrict__ b01p,
    const float* __restrict__ w10p, const float* __restrict__ b10p,
    const float* __restrict__ w12p, const float* __restrict__ b12p,
    const float* __restrict__ w21p, const float* __restrict__ b21p,
    const float* __restrict__ w23p, const float* __restrict__ b23p,
    const float* __restrict__ w32p, const float* __restrict__ b32p,
    float* __restrict__ out, int B) {
  // Wave-uniform parameter loads (lowered to s_load_b32 / KMcnt).
  P p;
  p.w01 = w01p[0]; p.b01 = b01p[0];
  p.w10 = w10p[0]; p.b10 = b10p[0];
  p.w12 = w12p[0]; p.b12 = b12p[0];
  p.w21 = w21p[0]; p.b21 = b21p[0];
  p.w23 = w23p[0]; p.b23 = b23p[0];
  p.w32 = w32p[0]; p.b32 = b32p[0];
  const float bs = (float)B;
  p.aRecBs = AREC * bs;
  p.recScale = 2.0f / bs;
  p.keep = 1.0f - BFW - LBW;
  p.oneMinusBfw = 1.0f - BFW;

  // Uniform output stream bases (return order: output, a0, a1n, a2n, a3n,
  // rec0, rec1, rec2). 8*B elements < 2^27, so 32-bit offsets suffice.
  const unsigned uB = (unsigned)B;
  float* __restrict__ oOut = out;
  float* __restrict__ oA0  = out + (size_t)uB;
  float* __restrict__ oA1  = out + (size_t)2 * uB;
  float* __restrict__ oA2  = out + (size_t)3 * uB;
  float* __restrict__ oA3  = out + (size_t)4 * uB;
  float* __restrict__ oR0  = out + (size_t)5 * uB;
  float* __restrict__ oR1  = out + (size_t)6 * uB;
  float* __restrict__ oR2  = out + (size_t)7 * uB;

  const unsigned i = (blockIdx.x * blockDim.x + threadIdx.x) * 4u;
  if (i >= uB) return;

  if (i + 4u <= uB) {
    // Fast path: 128-bit non-temporal loads/stores (touch-once streaming).
    v4f x0 = __builtin_nontemporal_load((const v4f*)(a0p + i));
    v4f x1 = __builtin_nontemporal_load((const v4f*)(a1p + i));
    v4f x2 = __builtin_nontemporal_load((const v4f*)(a2p + i));
    v4f x3 = __builtin_nontemporal_load((const v4f*)(a3p + i));

    v4f o, y1, y2, y3, r0, r1, r2;
#pragma unroll
    for (int k = 0; k < 4; ++k) {
      R r = compute_one(x0[k], x1[k], x2[k], x3[k], p);
      o[k] = r.outv;
      y1[k] = r.a1n; y2[k] = r.a2n; y3[k] = r.a3n;
      r0[k] = r.rec0; r1[k] = r.rec1; r2[k] = r.rec2;
    }

    __builtin_nontemporal_store(o,  (v4f*)(oOut + i));
    __builtin_nontemporal_store(x0, (v4f*)(oA0 + i));
    __builtin_nontemporal_store(y1, (v4f*)(oA1 + i));
    __builtin_nontemporal_store(y2, (v4f*)(oA2 + i));
    __builtin_nontemporal_store(y3, (v4f*)(oA3 + i));
    __builtin_nontemporal_store(r0, (v4f*)(oR0 + i));
    __builtin_nontemporal_store(r1, (v4f*)(oR1 + i));
    __builtin_nontemporal_store(r2, (v4f*)(oR2 + i));
  } else {
    // Tail (dead for B = 2^23; kept for generality).
    for (unsigned j = i; j < uB; ++j) {
      const float a0 = a0p[j], a1 = a1p[j], a2 = a2p[j], a3 = a3p[j];
      R r = compute_one(a0, a1, a2, a3, p);
      oOut[j] = r.outv;
      oA0[j] = a0;
      oA1[j] = r.a1n;
      oA2[j] = r.a2n;
      oA3[j] = r.a3n;
      oR0[j] = r.rec0;
      oR1[j] = r.rec1;
      oR2[j] = r.rec2;
    }
  }
}

extern "C" void kernel_launch(void* const* d_in, const int* in_sizes, int n_in,
                              void* d_out, int out_size, void* d_ws,
                              size_t ws_size, hipStream_t stream) {
  (void)n_in; (void)out_size; (void)d_ws; (void)ws_size;

  // setup_inputs() order: a0,a1,a2,a3, then (w01,b01),(w10,b10),(w12,b12),
  // (w21,b21),(w23,b23),(w32,b32)
  const float* a0 = (const float*)d_in[0];
  const float* a1 = (const float*)d_in[1];
  const float* a2 = (const float*)d_in[2];
  const float* a3 = (const float*)d_in[3];
  const float* w01 = (const float*)d_in[4];
  const float* b01 = (const float*)d_in[5];
  const float* w10 = (const float*)d_in[6];
  const float* b10 = (const float*)d_in[7];
  const float* w12 = (const float*)d_in[8];
  const float* b12 = (const float*)d_in[9];
  const float* w21 = (const float*)d_in[10];
  const float* b21 = (const float*)d_in[11];
  const float* w23 = (const float*)d_in[12];
  const float* b23 = (const float*)d_in[13];
  const float* w32 = (const float*)d_in[14];
  const float* b32 = (const float*)d_in[15];

  const int B = in_sizes[0];
  const long nThreads = ((long)B + 3) / 4;
  const int block = 256;  // 8 wave32s
  const long grid = (nThreads + block - 1) / block;

  unidim_kernel<<<(unsigned)grid, block, 0, stream>>>(
      a0, a1, a2, a3, w01, b01, w10, b10, w12, b12, w21, b21, w23, b23, w32,
      b32, (float*)d_out, B);
}